// IE_LS_49950469653059
// MI455X (gfx1250) — compile-verified
//
#include <hip/hip_runtime.h>
#include <cstdint>

// ---------------------------------------------------------------------------
// Lorenz Taylor-step, elementwise over 4,194,304 float3 points.
// Memory-bound (24 B/pt, ~110 FLOP/pt, AI ~4.6) -> optimize the HBM path:
//   global --async DMA--> LDS --ds_load_b128--> VGPR --compute--> LDS --async--> global
// Uses CDNA5 gfx1250 async-to-LDS instructions (ASYNCcnt) for perfectly
// coalesced b128 streaming of the awkward 12-byte AoS layout.
// ---------------------------------------------------------------------------

#define TPB        256u
#define PTS_TILE   1024u      // 256 threads * 4 points
#define TILE_F4    768u       // 1024 pts * 3 floats / 4 floats-per-float4
#define TILE_BYTES 12288u

typedef __attribute__((address_space(3))) float lds_f;

__device__ __forceinline__ void lorenz_point(float i0, float i1, float i2,
                                             float& o0, float& o1, float& o2) {
    const float B  = 8.0f / 3.0f;
    const float c1 = 0.005f;          // a0*h          = h/2
    const float c2 = 1.66666672e-5f;  // a1*h^2=a2*h^2 = h^2/6
    const float c4 = 1.25e-7f;        // a4*h^3        = 3h^3/24
    const float c5 = 4.16666679e-8f;  // a5*h^3=a6*h^3 = h^3/24

    float y0 = 10.0f * i0, y1 = 10.0f * i1, y2 = 10.0f * i2;

    // f(y)
    float f0 = 10.0f * (y1 - y0);
    float f1 = 28.0f * y0 - y0 * y2 - y1;
    float f2 = y0 * y1 - B * y2;

    // df*f
    float dff0 = 10.0f * (f1 - f0);
    float dff1 = (28.0f - y2) * f0 - f1 - y0 * f2;
    float dff2 = y1 * f0 + y0 * f1 - B * f2;

    // df*df*f
    float dfdff0 = 10.0f * (dff1 - dff0);
    float dfdff1 = (28.0f - y2) * dff0 - dff1 - y0 * dff2;
    float dfdff2 = y1 * dff0 + y0 * dff1 - B * dff2;

    // ddf(f,f)  (component 0 is identically zero)
    float ddfff1 = -2.0f * f0 * f2;
    float ddfff2 =  2.0f * f0 * f1;

    // ddf(dff,f) (component 0 is identically zero)
    float ddfdfff1 = -dff0 * f2 - dff2 * f0;
    float ddfdfff2 =  dff0 * f1 + dff1 * f0;

    // df*ddf(f,f), with ddfff0 == 0 folded out (keeps reference's y0*f1 term)
    float dfddfff0 = 10.0f * ddfff1;
    float dfddfff1 = -ddfff1 - y0 * ddfff2;
    float dfddfff2 = y0 * f1 - B * ddfff2;

    // df*df*df*f
    float dfdfdff0 = 10.0f * (dfdff1 - dfdff0);
    float dfdfdff1 = (28.0f - y2) * dfdff0 - dfdff1 - y0 * dfdff2;
    float dfdfdff2 = y1 * dfdff0 + y0 * dfdff1 - B * dfdff2;

    float r0 = f0 + c1 * dff0 + c2 * dfdff0
                  + c5 * (dfddfff0 + dfdfdff0);
    float r1 = f1 + c1 * dff1 + c2 * (ddfff1 + dfdff1)
                  + c4 * ddfdfff1 + c5 * (dfddfff1 + dfdfdff1);
    float r2 = f2 + c1 * dff2 + c2 * (ddfff2 + dfdff2)
                  + c4 * ddfdfff2 + c5 * (dfddfff2 + dfdfdff2);

    o0 = 0.1f * r0;  o1 = 0.1f * r1;  o2 = 0.1f * r2;
}

__global__ __launch_bounds__(256)
void lorenz_tile_kernel(const float* __restrict__ in, float* __restrict__ out) {
    __shared__ __align__(16) float tile[TILE_F4 * 4];   // 12 KB

    const unsigned t       = threadIdx.x;
    const unsigned tb      = blockIdx.x * TILE_BYTES;   // tile base (bytes), < 2^31
    const unsigned ldsBase = (unsigned)(unsigned long long)(lds_f*)tile;

    const unsigned long long gin  = (unsigned long long)in;
    const unsigned long long gout = (unsigned long long)out;

    // ---- Phase 1: async DMA global -> LDS, contiguous b128 per lane -------
#pragma unroll
    for (unsigned k = 0; k < 3; ++k) {
        unsigned goff = tb + (k * TPB + t) * 16u;       // global byte offset
        unsigned loff = ldsBase + (k * TPB + t) * 16u;  // LDS byte address
        asm volatile("global_load_async_to_lds_b128 %0, %1, %2"
                     :: "v"(loff), "v"(goff), "s"(gin) : "memory");
    }
    asm volatile("s_wait_asynccnt 0" ::: "memory");
    __syncthreads();

    // ---- Phase 2: compute 4 points per thread, in place in LDS ------------
    float4* tf = reinterpret_cast<float4*>(tile);
    float4 p0 = tf[3u * t + 0u];
    float4 p1 = tf[3u * t + 1u];
    float4 p2 = tf[3u * t + 2u];

    float a0, a1, a2, b0, b1, b2, q0, q1, q2, s0, s1, s2;
    lorenz_point(p0.x, p0.y, p0.z, a0, a1, a2);   // point 0
    lorenz_point(p0.w, p1.x, p1.y, b0, b1, b2);   // point 1
    lorenz_point(p1.z, p1.w, p2.x, q0, q1, q2);   // point 2
    lorenz_point(p2.y, p2.z, p2.w, s0, s1, s2);   // point 3

    tf[3u * t + 0u] = make_float4(a0, a1, a2, b0);
    tf[3u * t + 1u] = make_float4(b1, b2, q0, q1);
    tf[3u * t + 2u] = make_float4(q2, s0, s1, s2);
    __syncthreads();

    // ---- Phase 3: async DMA LDS -> global, contiguous b128 per lane -------
#pragma unroll
    for (unsigned k = 0; k < 3; ++k) {
        unsigned goff = tb + (k * TPB + t) * 16u;
        unsigned loff = ldsBase + (k * TPB + t) * 16u;
        asm volatile("global_store_async_from_lds_b128 %0, %1, %2"
                     :: "v"(goff), "v"(loff), "s"(gout) : "memory");
    }
    asm volatile("s_wait_asynccnt 0" ::: "memory");
}

// Scalar tail for any residue points (none for the 4,194,304-point harness).
__global__ __launch_bounds__(256)
void lorenz_tail_kernel(const float* __restrict__ in, float* __restrict__ out,
                        int start, int npts) {
    int i = start + (int)(blockIdx.x * blockDim.x + threadIdx.x);
    if (i < npts) {
        float o0, o1, o2;
        lorenz_point(in[3 * i + 0], in[3 * i + 1], in[3 * i + 2], o0, o1, o2);
        out[3 * i + 0] = o0;
        out[3 * i + 1] = o1;
        out[3 * i + 2] = o2;
    }
}

extern "C" void kernel_launch(void* const* d_in, const int* in_sizes, int n_in,
                              void* d_out, int out_size, void* d_ws, size_t ws_size,
                              hipStream_t stream) {
    (void)n_in; (void)out_size; (void)d_ws; (void)ws_size;
    const float* in  = (const float*)d_in[0];
    float*       out = (float*)d_out;

    long long nfloat = in_sizes[0];          // 12,582,912
    long long npts   = nfloat / 3;           //  4,194,304
    long long ntiles = npts / (long long)PTS_TILE;   // 4096
    long long rem    = npts - ntiles * (long long)PTS_TILE;

    if (ntiles > 0) {
        lorenz_tile_kernel<<<dim3((unsigned)ntiles), dim3(TPB), 0, stream>>>(in, out);
    }
    if (rem > 0) {
        unsigned blocks = (unsigned)((rem + TPB - 1) / TPB);
        lorenz_tail_kernel<<<dim3(blocks), dim3(TPB), 0, stream>>>(
            in, out, (int)(ntiles * (long long)PTS_TILE), (int)npts);
    }
}